// _StateSpaceLiteLayer_14542759264536
// MI455X (gfx1250) — compile-verified
//
#include <hip/hip_runtime.h>
#include <math.h>

typedef float v2f __attribute__((ext_vector_type(2)));
typedef float v8f __attribute__((ext_vector_type(8)));

#define B_   64
#define T_   512
#define D_   512
#define H_   1024
#define NBLK 64

__global__ void init_kernel(float* __restrict__ s0, unsigned* __restrict__ bar) {
    int i = blockIdx.x * blockDim.x + threadIdx.x;
    if (i < B_ * H_) s0[i] = 0.0f;
    if (i == 0) *bar = 0u;
}

// Device-wide barrier: monotonic counter, release on arrive, acquire on spin.
// Safe because all NBLK=64 blocks are co-resident on MI455X.
__device__ __forceinline__ void grid_sync(unsigned* bar, unsigned target) {
    __syncthreads();
    if (threadIdx.x == 0) {
        __threadfence();                       // release prior global writes (DEV scope)
        atomicAdd(bar, 1u);
        while (__hip_atomic_load(bar, __ATOMIC_ACQUIRE, __HIP_MEMORY_SCOPE_AGENT) < target) {
            __builtin_amdgcn_s_sleep(2);
        }
    }
    __syncthreads();
}

// Persistent scan kernel. Grid: 64 WGs = 4 batch tiles (M=16) x 16 H tiles (N=64).
// Block: 256 threads = 8 waves. Waves 0-3 and 4-7 split the K dimension
// (1024 state-K + 512 input-K) in half; accumulators are reduced through LDS.
// After the per-step grid barrier, WG `wg` does LayerNorm of state row `wg`.
__global__ void __launch_bounds__(256)
scan_kernel(const float* __restrict__ x,
            const float* __restrict__ W_in,
            const float* __restrict__ b_in,
            const float* __restrict__ W_state,
            const float* __restrict__ b_state,
            const float* __restrict__ decay,
            const float* __restrict__ gamma,
            const float* __restrict__ beta,
            float* __restrict__ s0,
            float* __restrict__ s1,
            unsigned* __restrict__ bar,
            float* __restrict__ y)
{
    __shared__ float accbuf[4][32][8];   // K-split partial accumulators
    __shared__ float red0[256];
    __shared__ float red1[256];

    const int wg   = blockIdx.x;
    const int m0   = (wg >> 4) << 4;                 // batch tile base: 0,16,32,48
    const int tid  = threadIdx.x;
    const int lane = tid & 31;
    const int wave = tid >> 5;
    const int w4   = wave & 3;                       // N sub-tile within WG
    const int kg   = wave >> 2;                      // K group: 0 = low half, 1 = high half
    const int n0   = ((wg & 15) << 6) + (w4 << 4);   // H column tile base
    const int half = lane >> 4;                      // WMMA fragment half
    const int l15  = lane & 15;
    const int kb   = half << 1;                      // K sub-offset 0 or 2
    const int c    = n0 + l15;                       // this lane's H column

    // ---- loop-invariant scalars / pointers ----
    const float bias = b_in[c] + b_state[c];
    const float dh   = 1.0f / (1.0f + expf(-decay[c]));        // sigmoid(decay)

    const size_t aoff = (size_t)(m0 + l15) * H_ + kg * (H_ / 2) + kb;          // state A frag
    const float* Bs   = W_state + (size_t)c * H_ + kg * (H_ / 2) + kb;         // W_state B frag
    const float* Bi   = W_in    + (size_t)c * D_ + kg * (D_ / 2) + kb;         // W_in B frag
    const float* Xb   = x + (size_t)(m0 + l15) * T_ * D_ + kg * (D_ / 2) + kb; // x A frag base

    const float4 g4  = *(const float4*)(gamma + tid * 4);
    const float4 be4 = *(const float4*)(beta  + tid * 4);

    for (int t = 0; t < T_; ++t) {
        const float* sin  = (t & 1) ? s1 : s0;
        float*       sout = (t & 1) ? s0 : s1;

        const float* A = sin + aoff;
        const float* X = Xb + (size_t)t * D_;
        if (t + 1 < T_) __builtin_prefetch(Xb + (size_t)(t + 1) * D_, 0, 1);

        v8f acc = {0.f,0.f,0.f,0.f,0.f,0.f,0.f,0.f};

        // state @ W_state^T : this wave's K half = 512
        for (int k = 0; k < H_ / 2; k += 4) {
            v2f a = *(const v2f*)(A  + k);
            v2f b = *(const v2f*)(Bs + k);
            acc = __builtin_amdgcn_wmma_f32_16x16x4_f32(false, a, false, b,
                                                        (short)0, acc, false, false);
        }
        // x[:,t,:] @ W_in^T : this wave's K half = 256
        for (int k = 0; k < D_ / 2; k += 4) {
            v2f a = *(const v2f*)(X  + k);
            v2f b = *(const v2f*)(Bi + k);
            acc = __builtin_amdgcn_wmma_f32_16x16x4_f32(false, a, false, b,
                                                        (short)0, acc, false, false);
        }

        // ---- K-split reduction through LDS, then epilogue by waves 0-3 ----
        if (kg) *(v8f*)&accbuf[w4][lane][0] = acc;
        __syncthreads();
        if (!kg) {
            acc += *(const v8f*)&accbuf[w4][lane][0];
#pragma unroll
            for (int j = 0; j < 8; ++j) {
                const int   br     = m0 + j + (half << 3);   // batch row (D-frag layout)
                const float sold   = sin[(size_t)br * H_ + c];
                const float driven = tanhf(acc[j] + bias);
                sout[(size_t)br * H_ + c] = dh * sold + (1.0f - dh) * driven;
            }
        }

        grid_sync(bar, (unsigned)(t + 1) * NBLK);

        // ---- LayerNorm of state row `wg`, write y[wg][t][:] ----
        const float4 v = *(const float4*)(sout + (size_t)wg * H_ + tid * 4);
        red0[tid] = v.x + v.y + v.z + v.w;
        red1[tid] = v.x*v.x + v.y*v.y + v.z*v.z + v.w*v.w;
        __syncthreads();
        for (int off = 128; off > 0; off >>= 1) {
            if (tid < off) { red0[tid] += red0[tid + off]; red1[tid] += red1[tid + off]; }
            __syncthreads();
        }
        const float mean = red0[0] * (1.0f / H_);
        const float var  = red1[0] * (1.0f / H_) - mean * mean;
        const float rstd = rsqrtf(var + 1e-5f);
        float4 o;
        o.x = (v.x - mean) * rstd * g4.x + be4.x;
        o.y = (v.y - mean) * rstd * g4.y + be4.y;
        o.z = (v.z - mean) * rstd * g4.z + be4.z;
        o.w = (v.w - mean) * rstd * g4.w + be4.w;
        *(float4*)(y + ((size_t)wg * T_ + t) * H_ + tid * 4) = o;
        // No extra barrier needed: next step writes only the dead ping-pong
        // buffer and re-reads sout; LDS reuse is guarded by grid_sync's
        // internal __syncthreads.
    }
}

extern "C" void kernel_launch(void* const* d_in, const int* in_sizes, int n_in,
                              void* d_out, int out_size, void* d_ws, size_t ws_size,
                              hipStream_t stream) {
    (void)in_sizes; (void)n_in; (void)out_size; (void)ws_size;
    const float* x       = (const float*)d_in[0];
    const float* W_in    = (const float*)d_in[1];
    const float* b_in    = (const float*)d_in[2];
    const float* W_state = (const float*)d_in[3];
    const float* b_state = (const float*)d_in[4];
    const float* decay   = (const float*)d_in[5];
    const float* gamma   = (const float*)d_in[6];
    const float* beta    = (const float*)d_in[7];
    float* y = (float*)d_out;

    float*    s0  = (float*)d_ws;                     // 64*1024 f32
    float*    s1  = s0 + B_ * H_;                     // 64*1024 f32
    unsigned* bar = (unsigned*)(s1 + B_ * H_);        // barrier counter

    init_kernel<<<(B_ * H_ + 255) / 256, 256, 0, stream>>>(s0, bar);
    scan_kernel<<<NBLK, 256, 0, stream>>>(x, W_in, b_in, W_state, b_state,
                                          decay, gamma, beta, s0, s1, bar, y);
}